// TS_DCM_70987219469024
// MI455X (gfx1250) — compile-verified
//
#include <hip/hip_runtime.h>
#include <hip/hip_bf16.h>
#include <math.h>

typedef __attribute__((ext_vector_type(2))) float v2f;
typedef __attribute__((ext_vector_type(8))) float v8f;
typedef __attribute__((ext_vector_type(4))) int   v4i;

#define NB     400          // (5*5 + 5*5) * 8 frames
#define DIMN   384
#define NP     196
#define NFR    8
#define WAY    5
#define SHOT   5
#define QPC    5
#define NSUP   200          // way*shot*F
#define NQRY   200
#define RTRI   73920        // 384*385/2
#define NTILE  24           // 384/16
#define NBLK   6            // 384/64 strip blocks
#define LDSW   72           // padded LDS row stride (floats); 2*72 % 64 == 16 -> disjoint banks
#define EPSC   1e-8f

// ---- workspace layout (floats) ----
#define OFF_W       ((size_t)0)                       // 400*196   = 78400
#define OFF_DIAG    ((size_t)78400)                   // 400*384   = 153600
#define OFF_ROWSUM  ((size_t)232000)                  // 400*384   = 153600
#define OFF_GSUM    ((size_t)385600)                  // 400
#define OFF_TSM     ((size_t)386000)                  // 25*64     = 1600
#define OFF_NORMP   ((size_t)387600)                  // 40
#define OFF_NORMQ   ((size_t)387640)                  // 200
#define OFF_AD      ((size_t)387840)                  // 200*40    = 8000
#define OFF_ZPROTO  ((size_t)395840)                  // 40*73920  = 2956800

#if defined(__HIP_DEVICE_COMPILE__) && __has_builtin(__builtin_amdgcn_global_load_async_to_lds_b128)
#define HAVE_ASYNC 1
#else
#define HAVE_ASYNC 0
#endif

#define ASP1 __attribute__((address_space(1)))
#define ASP3 __attribute__((address_space(3)))

// ---------------------------------------------------------------------------
__device__ __forceinline__ void wait_async() {
#if HAVE_ASYNC
#if __has_builtin(__builtin_amdgcn_s_wait_asynccnt)
    __builtin_amdgcn_s_wait_asynccnt(0);
#else
    asm volatile("s_wait_asynccnt 0" ::: "memory");
#endif
#endif
}

// copy one 196x64 strip of the patch matrix into LDS (row stride LDSW floats)
__device__ __forceinline__ void stage_strip(const float* __restrict__ patch, int strip,
                                            float* __restrict__ lds, int tid) {
    for (int t = tid; t < NP * 16; t += 256) {
        const int m = t >> 4;
        const int g = (t & 15) << 2;                 // 4-float group
        const float* src = patch + (size_t)m * DIMN + strip + g;
        float* dst = lds + m * LDSW + g;
#if HAVE_ASYNC
        __builtin_amdgcn_global_load_async_to_lds_b128((ASP1 v4i*)src, (ASP3 v4i*)dst, 0, 0);
#else
        dst[0] = src[0]; dst[1] = src[1]; dst[2] = src[2]; dst[3] = src[3];
#endif
    }
}

// ---------------------------------------------------------------------------
// zero rowsum + gsum regions
__global__ void k_zero(float* __restrict__ ws) {
    size_t i = (size_t)blockIdx.x * blockDim.x + threadIdx.x;
    size_t n = (OFF_GSUM + NB) - OFF_ROWSUM;   // 154000
    if (i < n) ws[OFF_ROWSUM + i] = 0.0f;
}

// ---------------------------------------------------------------------------
// per-batch: cls copy, cls norm, per-patch cosine weight w, diag of x2
__global__ void k_prep(const float* __restrict__ lsn, float* __restrict__ ws,
                       float* __restrict__ out_cls) {
    const int b = blockIdx.x;
    const float* cls   = lsn + (size_t)b * 197 * DIMN;
    const float* patch = cls + DIMN;
    __shared__ float sred[8];
    __shared__ float sncls;
    __shared__ float sw[NP];
    const int tid = threadIdx.x;          // 256 threads
    const int lane = tid & 31, wv = tid >> 5;

    float s = 0.0f;
    for (int d = tid; d < DIMN; d += 256) {
        float v = cls[d];
        s += v * v;
        out_cls[(size_t)b * DIMN + d] = v;
    }
    for (int m = 16; m; m >>= 1) s += __shfl_xor(s, m, 32);
    if (lane == 0) sred[wv] = s;
    __syncthreads();
    if (tid == 0) {
        float t = 0.0f;
        for (int i = 0; i < 8; i++) t += sred[i];
        sncls = fmaxf(sqrtf(t), EPSC);
    }
    __syncthreads();
    const float ncls = sncls;

    for (int p = wv; p < NP; p += 8) {
        const float* pp = patch + (size_t)p * DIMN;
        float dt = 0.0f, nr = 0.0f;
        for (int d = lane; d < DIMN; d += 32) {
            float pv = pp[d];
            dt += cls[d] * pv;
            nr += pv * pv;
        }
        for (int m = 16; m; m >>= 1) {
            dt += __shfl_xor(dt, m, 32);
            nr += __shfl_xor(nr, m, 32);
        }
        if (lane == 0) {
            float wval = dt / (ncls * fmaxf(sqrtf(nr), EPSC));
            sw[p] = wval;
            ws[OFF_W + (size_t)b * NP + p] = wval;
        }
    }
    __syncthreads();

    float* diag = ws + OFF_DIAG;
    for (int d = tid; d < DIMN; d += 256) {
        float acc = 0.0f;
        for (int m = 0; m < NP; m++) {
            float x = sw[m] * patch[(size_t)m * DIMN + d];
            acc += x * x;
        }
        diag[(size_t)b * DIMN + d] = acc * (1.0f / (2.0f * (float)NP));
    }
}

// ---------------------------------------------------------------------------
// one 16x16 Gram tile from LDS-staged strips via V_WMMA_F32_16X16X4_F32.
// fp32 A/B fragment layout: lanes 0-15 rows, K={0,1}; lanes 16-31 rows, K={2,3}
__device__ __forceinline__ v8f gram_tile_lds(const float* __restrict__ As,
                                             const float* __restrict__ Bs,
                                             int ti_loc, int tj_loc, int lane) {
    const int hi = lane >> 4, ln = lane & 15;
    const int ca = ti_loc * 16 + ln;
    const int cb = tj_loc * 16 + ln;
    v8f c = {0.f, 0.f, 0.f, 0.f, 0.f, 0.f, 0.f, 0.f};
    for (int k = 0; k < NP; k += 4) {
        const int m0 = k + 2 * hi;
        v2f a, bb;
        a.x  = As[m0 * LDSW + ca];
        a.y  = As[(m0 + 1) * LDSW + ca];
        bb.x = Bs[m0 * LDSW + cb];
        bb.y = Bs[(m0 + 1) * LDSW + cb];
        c = __builtin_amdgcn_wmma_f32_16x16x4_f32(false, a, false, bb,
                                                  (short)0, c, false, false);
    }
    return c;
}

__device__ __forceinline__ float dcov_elem(float diag_r, float diag_c, float gram) {
    float x2 = gram * (1.0f / (2.0f * (float)NP));
    float v = diag_r + diag_c - 2.0f * x2;
    return powf(fmaxf(v, 0.0f) + 1e-5f, 0.4f);
}

// stage both strips, apply w[m] in LDS
__device__ __forceinline__ void stage_block(const float* __restrict__ patch,
                                            const float* __restrict__ wb,
                                            int bi, int bj,
                                            float* As, float* Bs, float* sw,
                                            int tid) {
    stage_strip(patch, bi * 64, As, tid);
    stage_strip(patch, bj * 64, Bs, tid);
    if (tid < NP) sw[tid] = wb[tid];
    wait_async();
    __syncthreads();
    for (int t = tid; t < NP * 64; t += 256) {
        const int m = t >> 6, j = t & 63;
        const float wv = sw[m];
        As[m * LDSW + j] *= wv;
        Bs[m * LDSW + j] *= wv;
    }
    __syncthreads();
}

// pass 1: all 36 (64x64) blocks per batch; accumulate dcov row sums
__global__ void __launch_bounds__(256) k_gram_rowsum(const float* __restrict__ lsn,
                                                     float* __restrict__ ws) {
    __shared__ float As[NP * LDSW];
    __shared__ float Bs[NP * LDSW];
    __shared__ float sw[NP];
    __shared__ float rsum[64];
    const int idx = blockIdx.x;
    const int b  = idx / (NBLK * NBLK);
    const int r  = idx - b * (NBLK * NBLK);
    const int bi = r / NBLK, bj = r - bi * NBLK;
    const float* patch = lsn + (size_t)b * 197 * DIMN + DIMN;
    const float* wb    = ws + OFF_W    + (size_t)b * NP;
    const float* diag  = ws + OFF_DIAG + (size_t)b * DIMN;
    const int tid = threadIdx.x;
    if (tid < 64) rsum[tid] = 0.0f;

    stage_block(patch, wb, bi, bj, As, Bs, sw, tid);

    const int lane = tid & 31, wv = tid >> 5;
    const int hi = lane >> 4, ln = lane & 15;
    for (int t16 = wv; t16 < 16; t16 += 8) {
        const int ti_loc = t16 >> 2, tj_loc = t16 & 3;
        v8f c = gram_tile_lds(As, Bs, ti_loc, tj_loc, lane);
        const float dcol = diag[bj * 64 + tj_loc * 16 + ln];
        #pragma unroll
        for (int rr = 0; rr < 8; rr++) {
            const int lr = ti_loc * 16 + rr + 8 * hi;
            float v = dcov_elem(diag[bi * 64 + lr], dcol, c[rr]);
            for (int m = 1; m < 16; m <<= 1) v += __shfl_xor(v, m, 32);
            if (ln == 0) atomicAdd(&rsum[lr], v);           // ds_add_f32
        }
    }
    __syncthreads();
    if (tid < 64)
        atomicAdd(&ws[OFF_ROWSUM + (size_t)b * DIMN + bi * 64 + tid], rsum[tid]);
}

// gsum[b] = sum over 384 row sums
__global__ void k_gsum(float* __restrict__ ws) {
    const int b = blockIdx.x;
    const float* rs = ws + OFF_ROWSUM + (size_t)b * DIMN;
    __shared__ float sred[4];
    const int tid = threadIdx.x;      // 128
    float s = 0.0f;
    for (int d = tid; d < DIMN; d += 128) s += rs[d];
    for (int m = 16; m; m >>= 1) s += __shfl_xor(s, m, 32);
    if ((tid & 31) == 0) sred[tid >> 5] = s;
    __syncthreads();
    if (tid == 0) ws[OFF_GSUM + b] = sred[0] + sred[1] + sred[2] + sred[3];
}

// pass 2: 21 upper blocks per batch; write centered triu values
__global__ void __launch_bounds__(256) k_gram_center(const float* __restrict__ lsn,
                                                     const float* __restrict__ ws,
                                                     float* __restrict__ out_adc) {
    __shared__ float As[NP * LDSW];
    __shared__ float Bs[NP * LDSW];
    __shared__ float sw[NP];
    const int idx = blockIdx.x;
    const int b = idx / 21;
    int p = idx - b * 21;
    int bi = 0;
    while (p >= (NBLK - bi)) { p -= (NBLK - bi); bi++; }
    const int bj = bi + p;

    const float* patch  = lsn + (size_t)b * 197 * DIMN + DIMN;
    const float* wb     = ws + OFF_W      + (size_t)b * NP;
    const float* diag   = ws + OFF_DIAG   + (size_t)b * DIMN;
    const float* rowsum = ws + OFF_ROWSUM + (size_t)b * DIMN;
    const float gm = ws[OFF_GSUM + b] * (1.0f / ((float)DIMN * (float)DIMN));
    const int tid = threadIdx.x;

    stage_block(patch, wb, bi, bj, As, Bs, sw, tid);

    const int lane = tid & 31, wv = tid >> 5;
    const int hi = lane >> 4, ln = lane & 15;
    for (int t16 = wv; t16 < 16; t16 += 8) {
        const int ti_loc = t16 >> 2, tj_loc = t16 & 3;
        if (bi == bj && ti_loc > tj_loc) continue;   // wave-uniform skip
        v8f c = gram_tile_lds(As, Bs, ti_loc, tj_loc, lane);
        const int gc = bj * 64 + tj_loc * 16 + ln;
        const float dcol = diag[gc];
        const float cm_c = rowsum[gc] * (1.0f / (float)DIMN);
        #pragma unroll
        for (int rr = 0; rr < 8; rr++) {
            const int gr = bi * 64 + ti_loc * 16 + rr + 8 * hi;
            float v = dcov_elem(diag[gr], dcol, c[rr]);
            v = v - rowsum[gr] * (1.0f / (float)DIMN) - cm_c + gm;
            if (gr <= gc) {
                int oidx = gr * DIMN - (gr * (gr - 1)) / 2 + (gc - gr);
                out_adc[(size_t)b * RTRI + oidx] = v;
            }
        }
    }
}

// ---------------------------------------------------------------------------
// task prototypes + tsm[q][f] = <task_query[q]+task_support, gen_weight[f]>
__global__ void k_tsm(const float* __restrict__ out_cls, const float* __restrict__ gw,
                      float* __restrict__ ws) {
    __shared__ float ts[DIMN];
    __shared__ float tq[25 * DIMN];
    const int tid = threadIdx.x;      // 256
    for (int d = tid; d < DIMN; d += 256) {
        float s = 0.0f;
        for (int i = 0; i < NSUP; i++) s += out_cls[(size_t)i * DIMN + d];
        ts[d] = s * (1.0f / (float)NSUP);
    }
    for (int e = tid; e < 25 * DIMN; e += 256) {
        const int q = e / DIMN, d = e - q * DIMN;
        float s = 0.0f;
        for (int f = 0; f < NFR; f++)
            s += out_cls[(size_t)(NSUP + q * NFR + f) * DIMN + d];
        tq[e] = s * (1.0f / (float)NFR);
    }
    __syncthreads();
    for (int e = tid; e < 25 * 64; e += 256) {
        const int q = e / 64, f = e - q * 64;
        float s = 0.0f;
        for (int d = 0; d < DIMN; d++)
            s += (tq[q * DIMN + d] + ts[d]) * gw[(size_t)f * DIMN + d];
        ws[OFF_TSM + e] = s;
    }
}

// z_proto[c*8+f] = mean over shot of adc[(c*5+s)*8+f]
__global__ void k_zproto(const float* __restrict__ out_adc, float* __restrict__ ws) {
    const int row = blockIdx.y;                 // 0..39
    const int c = row >> 3, f = row & 7;
    const int r = blockIdx.x * 256 + threadIdx.x;
    if (r >= RTRI) return;
    float s = 0.0f;
    for (int sh = 0; sh < SHOT; sh++) {
        const int samp = (c * SHOT + sh) * NFR + f;
        s += out_adc[(size_t)samp * RTRI + r];
    }
    ws[OFF_ZPROTO + (size_t)row * RTRI + r] = s * (1.0f / (float)SHOT);
}

// L2 norms of proto (40) and query (200) rows
__global__ void k_norms(const float* __restrict__ out_adc, float* __restrict__ ws) {
    const int row = blockIdx.x;                 // 0..239
    const float* src = (row < 40)
        ? (ws + OFF_ZPROTO + (size_t)row * RTRI)
        : (out_adc + (size_t)(NSUP + row - 40) * RTRI);
    __shared__ float sred[8];
    const int tid = threadIdx.x;                // 256
    float s = 0.0f;
    for (int r = tid; r < RTRI; r += 256) { float v = src[r]; s += v * v; }
    for (int m = 16; m; m >>= 1) s += __shfl_xor(s, m, 32);
    if ((tid & 31) == 0) sred[tid >> 5] = s;
    __syncthreads();
    if (tid == 0) {
        float t = 0.0f;
        for (int i = 0; i < 8; i++) t += sred[i];
        float n = fmaxf(sqrtf(t), EPSC);
        if (row < 40) ws[OFF_NORMP + row] = n;
        else          ws[OFF_NORMQ + row - 40] = n;
    }
}

// ad[q][p] = <query_q, proto_p> / (nq*np)
__global__ void k_ad(const float* __restrict__ out_adc, float* __restrict__ ws) {
    const int q = blockIdx.x, p = blockIdx.y;
    const float* qr = out_adc + (size_t)(NSUP + q) * RTRI;
    const float* pr = ws + OFF_ZPROTO + (size_t)p * RTRI;
    __shared__ float sred[8];
    const int tid = threadIdx.x;                // 256
    float s = 0.0f;
    for (int r = tid; r < RTRI; r += 256) s += qr[r] * pr[r];
    for (int m = 16; m; m >>= 1) s += __shfl_xor(s, m, 32);
    if ((tid & 31) == 0) sred[tid >> 5] = s;
    __syncthreads();
    if (tid == 0) {
        float t = 0.0f;
        for (int i = 0; i < 8; i++) t += sred[i];
        ws[OFF_AD + q * 40 + p] = t / (ws[OFF_NORMQ + q] * ws[OFF_NORMP + p]);
    }
}

// score[q][w] = sum_{f1,f2} ad[q*8+f1][w*8+f2] * tsm[q][f1*8+f2]
__global__ void k_score(const float* __restrict__ ws, float* __restrict__ out_score) {
    const int idx = threadIdx.x;
    if (idx >= 125) return;
    const int q = idx / WAY, w = idx - q * WAY;
    float s = 0.0f;
    for (int f1 = 0; f1 < NFR; f1++)
        for (int f2 = 0; f2 < NFR; f2++)
            s += ws[OFF_AD + (q * NFR + f1) * 40 + (w * NFR + f2)]
               * ws[OFF_TSM + q * 64 + f1 * NFR + f2];
    out_score[idx] = s;
}

// ---------------------------------------------------------------------------
extern "C" void kernel_launch(void* const* d_in, const int* in_sizes, int n_in,
                              void* d_out, int out_size, void* d_ws, size_t ws_size,
                              hipStream_t stream) {
    const float* lsn = (const float*)d_in[0];           // (400, 197, 384)
    const float* gw  = (const float*)d_in[1];           // (64, 384)
    (void)in_sizes; (void)n_in; (void)out_size; (void)ws_size;

    float* out       = (float*)d_out;
    float* out_score = out;                             // 125
    float* out_adc   = out + 125;                       // 400*73920
    float* out_cls   = out + 125 + (size_t)NB * RTRI;   // 400*384
    float* ws        = (float*)d_ws;

    k_zero<<<(154000 + 255) / 256, 256, 0, stream>>>(ws);
    k_prep<<<NB, 256, 0, stream>>>(lsn, ws, out_cls);

    // pass 1: 400 * 36 blocks of 64x64 (full matrix, row sums)
    k_gram_rowsum<<<NB * NBLK * NBLK, 256, 0, stream>>>(lsn, ws);
    k_gsum<<<NB, 128, 0, stream>>>(ws);
    // pass 2: 400 * 21 upper blocks (centered triu output)
    k_gram_center<<<NB * 21, 256, 0, stream>>>(lsn, ws, out_adc);

    k_tsm<<<1, 256, 0, stream>>>(out_cls, gw, ws);
    k_zproto<<<dim3((RTRI + 255) / 256, 40), 256, 0, stream>>>(out_adc, ws);
    k_norms<<<240, 256, 0, stream>>>(out_adc, ws);
    k_ad<<<dim3(NQRY, 40), 256, 0, stream>>>(out_adc, ws);
    k_score<<<1, 128, 0, stream>>>(ws, out_score);
}